// RawStack_39539468927785
// MI455X (gfx1250) — compile-verified
//
#include <hip/hip_runtime.h>

// ---------------------------------------------------------------------------
// Fused 10-layer dilated-conv stack (WaveNet RawBlocks), fp32.
// HBM traffic ~117 MB (async DMA in, b128 reg stores out) but the real limiter
// is LDS bank bandwidth, so compute is register-resident: each thread owns
// R=20 contiguous elements of (h,x); per layer only the dilation halo goes
// through LDS (compacted for d<R, full-position for d>=R). ~129 B/elem LDS
// traffic vs ~240 B/elem for a pure LDS ping-pong.
// ---------------------------------------------------------------------------

#define N_LAYERS 10
#define HALO     1023              // 1+2+4+...+512
#define TILE     4096              // outputs per workgroup
#define BLOCK    256               // 8 waves (wave32)
#define R        20                // elements per thread; 256*20 = 5120
#define LBUF     (TILE + HALO + 1) // 5120 valid positions
#define LPAD     (LBUF + 4)        // +4 pad for layer-0's [c+R] read

typedef float f4 __attribute__((ext_vector_type(4)));

#if __has_builtin(__builtin_amdgcn_tanhf)
#define TANHF(v) __builtin_amdgcn_tanhf(v)
#elif __has_builtin(__builtin_amdgcn_tanh_f32)
#define TANHF(v) __builtin_amdgcn_tanh_f32(v)
#else
#define TANHF(v) tanhf(v)
#endif

// copies 16 bytes from global memory at byte offset glb_byte_off past base to LDS byte address lds_byte   (ASYNCcnt DMA)
__device__ __forceinline__ void async_load_f4(uint32_t lds_byte,
                                              uint32_t glb_byte_off,
                                              const float* base) {
  asm volatile("global_load_async_to_lds_b128 %0, %1, %2"
               :: "v"(lds_byte), "v"(glb_byte_off), "s"(base)
               : "memory");
}

__device__ __forceinline__ void wait_async0() {
#if __has_builtin(__builtin_amdgcn_s_wait_asynccnt)
  __builtin_amdgcn_s_wait_asynccnt(0);
#else
  asm volatile("s_wait_asynccnt 0" ::: "memory");
#endif
}

// d < R: neighbor needs only my first D elements -> compact publish at tid*D.
template<int D>
__device__ __forceinline__ void layer_compact(float (&rh)[R], float (&rx)[R],
                                              float* Ph, float* Px, int tid,
                                              float h0, float h1,
                                              float x0, float x1) {
#pragma unroll
  for (int j = 0; j < D; ++j) { Ph[tid * D + j] = rh[j]; Px[tid * D + j] = rx[j]; }
  __syncthreads();
  float hh[D], hx[D];
#pragma unroll
  for (int j = 0; j < D; ++j) {           // tid==255 reads stale slot: halo
    hh[j] = Ph[(tid + 1) * D + j];        // region is garbage there anyway
    hx[j] = Px[(tid + 1) * D + j];
  }
#pragma unroll
  for (int t = 0; t < R; ++t) {           // ascending t: in-place is safe
    float hd = (t + D < R) ? rh[t + D] : hh[t + D - R];
    float xd = (t + D < R) ? rx[t + D] : hx[t + D - R];
    float hv = fmaf(h1, hd, h0 * rh[t]);
    float xv = fmaf(x1, xd, x0 * rx[t]);
    rh[t] = hv; rx[t] = TANHF(hv + xv);
  }
}

// d >= R: publish whole chunk at natural positions, halo-read [c+D, c+D+R).
template<int D>
__device__ __forceinline__ void layer_full(float (&rh)[R], float (&rx)[R],
                                           float* Ph, float* Px, int c,
                                           float h0, float h1,
                                           float x0, float x1) {
#pragma unroll
  for (int q = 0; q < R / 4; ++q) {
    *(f4*)&Ph[c + 4 * q] = *(const f4*)&rh[4 * q];
    *(f4*)&Px[c + 4 * q] = *(const f4*)&rx[4 * q];
  }
  __syncthreads();
  int hb = c + D;                          // clamp keeps reads in-allocation;
  if (hb > LBUF - R) hb = LBUF - R;        // clamped threads are garbage zone
  float hh[R], hx[R];
#pragma unroll
  for (int q = 0; q < R / 4; ++q) {
    *(f4*)&hh[4 * q] = *(const f4*)&Ph[hb + 4 * q];
    *(f4*)&hx[4 * q] = *(const f4*)&Px[hb + 4 * q];
  }
#pragma unroll
  for (int t = 0; t < R; ++t) {            // D >= R: in[t+D] always external
    float hv = fmaf(h1, hh[t], h0 * rh[t]);
    float xv = fmaf(x1, hx[t], x0 * rx[t]);
    rh[t] = hv; rx[t] = TANHF(hv + xv);
  }
}

__global__ __launch_bounds__(BLOCK)
void rawstack_fused_kernel(const float* __restrict__ h,
                           const float* __restrict__ x,
                           const float* __restrict__ wh,
                           const float* __restrict__ wx,
                           float* __restrict__ out,
                           int T, int outLen) {
  __shared__ __align__(16) float Ah[LPAD], Ax[LPAD], Bh[LPAD], Bx[LPAD];

  const int tile = blockIdx.x;
  const int row  = blockIdx.y;
  const int tid  = threadIdx.x;
  const int s0   = tile * TILE;
  const int c    = tid * R;                // chunk start position in tile

  const float* hrow = h + (size_t)row * (size_t)T;
  const float* xrow = x + (size_t)row * (size_t)T;

  // ---- async DMA in: [s0, s0+LBUF) of h and x -> LDS (A buffers) ----
  const uint32_t maxE4 = (uint32_t)(T / 4 - 1);
  const uint32_t s0_4  = (uint32_t)(s0 >> 2);
  const uint32_t ldsH  = (uint32_t)(size_t)(void*)Ah;
  const uint32_t ldsX  = (uint32_t)(size_t)(void*)Ax;
#pragma unroll
  for (int j4 = 0; j4 < LBUF / 4; j4 += BLOCK) {   // 5 passes, all lanes live
    uint32_t jj = (uint32_t)(j4 + tid);
    uint32_t e4 = s0_4 + jj;
    if (e4 > maxE4) e4 = maxE4;                    // clamp: garbage halo only
    async_load_f4(ldsH + jj * 16u, e4 * 16u, hrow);
    async_load_f4(ldsX + jj * 16u, e4 * 16u, xrow);
  }
  wait_async0();
  __syncthreads();

  // ---- layer 0 (d=1): inputs already "published" in A by the DMA ----
  float rh[R], rx[R];
  {
    float th[R + 4], tx[R + 4];
#pragma unroll
    for (int q = 0; q < R / 4; ++q) {
      *(f4*)&th[4 * q] = *(const f4*)&Ah[c + 4 * q];
      *(f4*)&tx[4 * q] = *(const f4*)&Ax[c + 4 * q];
    }
    th[R] = Ah[c + R];                     // one-past halo (LPAD covers tid 255)
    tx[R] = Ax[c + R];
    const float h0 = wh[0], h1 = wh[1], x0 = wx[0], x1 = wx[1];
#pragma unroll
    for (int j = 0; j < R; ++j) {
      float hv = fmaf(h1, th[j + 1], h0 * th[j]);
      float xv = fmaf(x1, tx[j + 1], x0 * tx[j]);
      rh[j] = hv; rx[j] = TANHF(hv + xv);
    }
  }

  // ---- layers 1..9: alternate publish buffers B,A,B,... (2-barrier WAR gap)
  layer_compact<2 >(rh, rx, Bh, Bx, tid, wh[2],  wh[3],  wx[2],  wx[3]);
  layer_compact<4 >(rh, rx, Ah, Ax, tid, wh[4],  wh[5],  wx[4],  wx[5]);
  layer_compact<8 >(rh, rx, Bh, Bx, tid, wh[6],  wh[7],  wx[6],  wx[7]);
  layer_compact<16>(rh, rx, Ah, Ax, tid, wh[8],  wh[9],  wx[8],  wx[9]);
  layer_full<32 >(rh, rx, Bh, Bx, c, wh[10], wh[11], wx[10], wx[11]);
  layer_full<64 >(rh, rx, Ah, Ax, c, wh[12], wh[13], wx[12], wx[13]);
  layer_full<128>(rh, rx, Bh, Bx, c, wh[14], wh[15], wx[14], wx[15]);
  layer_full<256>(rh, rx, Ah, Ax, c, wh[16], wh[17], wx[16], wx[17]);
  layer_full<512>(rh, rx, Bh, Bx, c, wh[18], wh[19], wx[18], wx[19]);

  // ---- store final x straight from registers (b128; 4B-aligned is legal) ----
  float* orow  = out + (size_t)row * (size_t)outLen;
  const int g0 = s0 + c;
  const int lim = outLen - s0;             // valid outputs in this tile
  if (c + R <= lim) {
#pragma unroll
    for (int q = 0; q < R / 4; ++q)
      *(f4*)&orow[g0 + 4 * q] = *(const f4*)&rx[4 * q];
  } else {
#pragma unroll
    for (int j = 0; j < R; ++j)
      if (c + j < lim) orow[g0 + j] = rx[j];
  }
}

extern "C" void kernel_launch(void* const* d_in, const int* in_sizes, int n_in,
                              void* d_out, int out_size, void* d_ws, size_t ws_size,
                              hipStream_t stream) {
  const float* h  = (const float*)d_in[0];   // (64, 1, 131072) f32
  const float* x  = (const float*)d_in[1];   // (64, 1, 131072) f32
  const float* wh = (const float*)d_in[2];   // (10, 2) f32
  const float* wx = (const float*)d_in[3];   // (10, 2) f32
  float* out = (float*)d_out;                // (64, 130049) f32

  const int B      = 64;
  const int T      = in_sizes[0] / B;        // 131072 (C == 1)
  const int outLen = T - HALO;               // 130049
  const int nt     = (outLen + TILE - 1) / TILE;  // 32 tiles

  dim3 grid(nt, B), block(BLOCK);
  rawstack_fused_kernel<<<grid, block, 0, stream>>>(h, x, wh, wx, out, T, outLen);
}